// GINModel_38010460569655
// MI455X (gfx1250) — compile-verified
//
#include <hip/hip_runtime.h>

typedef __attribute__((ext_vector_type(2))) float v2f;
typedef __attribute__((ext_vector_type(8))) float v8f;

#define HID 128
#define TILE_ROWS 64
#define LDS_STRIDE 132  // pad: 4m banks apart per row -> conflict-free b64 LDS reads

__global__ void embed_kernel(const int* __restrict__ feats,
                             const float* __restrict__ ke,
                             const float* __restrict__ ve,
                             float* __restrict__ h, int nNodes) {
  int i = blockIdx.x * blockDim.x + threadIdx.x;
  if (i >= nNodes * HID) return;
  int node = i >> 7;
  int j = i & (HID - 1);
  int f0 = feats[node * 2 + 0];
  int f1 = feats[node * 2 + 1];
  float v = ke[f0 * HID + j] + ve[f1 * HID + j];
  h[i] = v > 0.f ? v : 0.f;
}

__global__ void copy_kernel(const float4* __restrict__ a, float4* __restrict__ b, int n4) {
  int i = blockIdx.x * blockDim.x + threadIdx.x;
  if (i < n4) b[i] = a[i];
}

// One wave32 per edge: z[dst] += h[src], 4 float atomics per lane.
__global__ void scatter_add_kernel(const float* __restrict__ h,
                                   const int* __restrict__ src,
                                   const int* __restrict__ dst,
                                   float* __restrict__ z, int nEdges) {
  int gid = blockIdx.x * blockDim.x + threadIdx.x;
  int wid = gid >> 5;
  int lane = threadIdx.x & 31;
  if (wid >= nEdges) return;
  int s = src[wid];  // wave-uniform
  int d = dst[wid];
  const float* hs = h + (size_t)s * HID;
  float* zd = z + (size_t)d * HID;
#pragma unroll
  for (int j = 0; j < HID / 32; ++j)
    atomicAdd(&zd[lane + j * 32], hs[lane + j * 32]);
}

// Y[N x Co] = act(X[N x 128] @ W[Co x 128]^T + bias), fp32 via V_WMMA_F32_16X16X4_F32.
// 128 threads = 4 waves; block covers 64 rows; each wave owns a 16-row subtile
// and loops over 16-col tiles. EXEC is full at every WMMA.
__global__ void __launch_bounds__(128)
gemm_bias_act_kernel(const float* __restrict__ X, const float* __restrict__ W,
                     const float* __restrict__ bias, float* __restrict__ Y,
                     int N, int Co, int doRelu) {
  __shared__ float Xs[TILE_ROWS * LDS_STRIDE];
  const int row0 = blockIdx.x * TILE_ROWS;
  const int t = threadIdx.x;

  // Stage 64x128 X tile into LDS (zero-fill OOB rows), float4 coalesced.
  const float4* X4 = (const float4*)X;
#pragma unroll
  for (int i = 0; i < 16; ++i) {
    int idx = t + i * 128;  // 0..2047 float4 slots
    int r = idx >> 5;       // 32 float4 per row
    int c4 = idx & 31;
    float4 v = make_float4(0.f, 0.f, 0.f, 0.f);
    int gr = row0 + r;
    if (gr < N) v = X4[(size_t)gr * (HID / 4) + c4];
    *(float4*)&Xs[r * LDS_STRIDE + (c4 << 2)] = v;
  }
  __syncthreads();

  const int lane = t & 31;
  const int wave = t >> 5;              // 0..3 -> row subtile
  const int m16 = lane & 15;            // A row / B col / D col
  const int khalf = (lane >> 4) << 1;   // 0 or 2 (K offset for hi half-wave)
  const float* xrow = &Xs[(wave * 16 + m16) * LDS_STRIDE];

  const int nColTiles = (Co + 15) >> 4;
  for (int ct = 0; ct < nColTiles; ++ct) {
    const int gn = (ct << 4) + m16;
    const int ncl = gn < Co ? gn : Co - 1;     // clamp: padded cols recompute
    const float* wrow = &W[(size_t)ncl * HID]; // col Co-1, discarded at store
    v8f c = {};
#pragma unroll
    for (int k = 0; k < HID; k += 4) {
      v2f a = *(const v2f*)(xrow + k + khalf);  // ds_load_b64
      v2f b = *(const v2f*)(wrow + k + khalf);  // global_load_b64, L0-resident
      c = __builtin_amdgcn_wmma_f32_16x16x4_f32(false, a, false, b, (short)0, c,
                                                false, false);
    }
    float bv = 0.f;
    if (bias) bv = (gn < Co) ? bias[ncl] : 0.f;
#pragma unroll
    for (int v = 0; v < 8; ++v) {
      int gm = row0 + wave * 16 + v + ((lane >> 4) << 3);
      if (gm < N && gn < Co) {
        float y = c[v] + bv;
        if (doRelu) y = y > 0.f ? y : 0.f;
        Y[(size_t)gm * Co + gn] = y;
      }
    }
  }
}

extern "C" void kernel_launch(void* const* d_in, const int* in_sizes, int n_in,
                              void* d_out, int out_size, void* d_ws, size_t ws_size,
                              hipStream_t stream) {
  const int* feats = (const int*)d_in[0];
  const int* src   = (const int*)d_in[1];
  const int* dst   = (const int*)d_in[2];
  const float* ke  = (const float*)d_in[3];
  const float* ve  = (const float*)d_in[4];
  const float* W1  = (const float*)d_in[5];
  const float* b1  = (const float*)d_in[6];
  const float* W2  = (const float*)d_in[7];
  const float* b2  = (const float*)d_in[8];
  const float* Wc  = (const float*)d_in[9];
  float* out = (float*)d_out;

  const int nNodes  = in_sizes[0] / 2;
  const int nEdges  = in_sizes[1];
  const int nOut    = in_sizes[9] / HID;           // 40
  const int nLayers = in_sizes[5] / (HID * HID);   // 3

  size_t feat = (size_t)nNodes * HID;
  float* h  = (float*)d_ws;     // [nNodes][128]
  float* z  = h + feat;         // [nNodes][128]
  float* tb = z + feat;         // [nNodes][128]

  embed_kernel<<<(nNodes * HID + 255) / 256, 256, 0, stream>>>(feats, ke, ve, h, nNodes);

  const int n4 = (int)(feat / 4);
  const int gemmBlocks = (nNodes + TILE_ROWS - 1) / TILE_ROWS;
  const long long scatterThreads = (long long)nEdges * 32;
  const int scatterBlocks = (int)((scatterThreads + 255) / 256);

  for (int i = 0; i < nLayers; ++i) {
    // z = h  (so z = h + agg after scatter; GIN eps = 0)
    copy_kernel<<<(n4 + 255) / 256, 256, 0, stream>>>((const float4*)h, (float4*)z, n4);
    scatter_add_kernel<<<scatterBlocks, 256, 0, stream>>>(h, src, dst, z, nEdges);
    // tb = relu(z @ W1[i]^T + b1[i]);  h = tb @ W2[i]^T + b2[i]
    gemm_bias_act_kernel<<<gemmBlocks, 128, 0, stream>>>(
        z, W1 + (size_t)i * HID * HID, b1 + (size_t)i * HID, tb, nNodes, HID, 1);
    gemm_bias_act_kernel<<<gemmBlocks, 128, 0, stream>>>(
        tb, W2 + (size_t)i * HID * HID, b2 + (size_t)i * HID, h, nNodes, HID, 0);
  }
  // out = h @ Wc^T  (no bias)
  gemm_bias_act_kernel<<<gemmBlocks, 128, 0, stream>>>(
      h, Wc, nullptr, out, nNodes, nOut, 0);
}